// VolumetricTriangulation_30468497997961
// MI455X (gfx1250) — compile-verified
//
#include <hip/hip_runtime.h>
#include <hip/hip_bf16.h>

// Problem constants (match reference)
#define BB 2
#define CC 4
#define KK 32
#define HH 96
#define WW 96
#define DD 64
#define NN (DD * DD * DD)        // 262144 voxels per batch
#define HW (HH * WW)             // 9216

typedef __attribute__((ext_vector_type(2))) float v2f;
typedef __attribute__((ext_vector_type(8))) float v8f;

// Block: 256 threads = 64 voxels x 4 k-chunks (8 k each).
// Grid:  (NN/64, BB)
__global__ __launch_bounds__(256)
void VolumetricTriangulation_30468497997961_kernel(
    const float* __restrict__ heatmaps,   // (B,C,K,H,W)
    const float* __restrict__ Pm,         // (B,C,3,4)
    const float* __restrict__ coords,     // (B,D,D,D,3)
    const float* __restrict__ conf,       // (B,C,K)
    float* __restrict__ out)              // (B,K,D,D,D)
{
    __shared__ float sProj[64 * 16];      // 64 voxels x 16 proj columns (12 live)

    const int tid  = threadIdx.x;
    const int wave = tid >> 5;
    const int lane = tid & 31;
    const int b      = blockIdx.y;
    const int n_base = blockIdx.x * 64;

    // ---------------- Phase 1: WMMA projection ----------------
    // Waves 0..3 each handle a 16-voxel tile. One v_wmma_f32_16x16x4_f32 computes
    // D[m][n] = sum_k homo[m][k] * P[n/3][n%3][k] for all 4 cameras at once.
    if (wave < 4) {
        const int t     = wave;            // tile id
        const int mlane = lane & 15;       // M (voxel-in-tile) for A; N (column) for B/D
        const int n_g   = n_base + t * 16 + mlane;

        // A matrix 16x4 f32: lanes 0-15 hold K=0,1 (x,y); lanes 16-31 hold K=2,3 (z,1)
        const float* cc3 = coords + ((size_t)b * NN + n_g) * 3;
        v2f A;
        if (lane < 16) { A.x = cc3[0]; A.y = cc3[1]; }
        else           { A.x = cc3[2]; A.y = 1.0f;  }

        // B matrix 4x16 f32: column n = 3*cam + comp carries P[b][cam][comp][k];
        // lanes 0-15 hold rows K=0,1; lanes 16-31 hold rows K=2,3 (mirrors A layout).
        v2f Bm; Bm.x = 0.0f; Bm.y = 0.0f;
        if (mlane < 12) {
            const int cam = mlane / 3, comp = mlane % 3;
            const float* Pp = Pm + (size_t)(b * CC + cam) * 12 + comp * 4;
            const int k0 = (lane < 16) ? 0 : 2;
            Bm.x = Pp[k0];
            Bm.y = Pp[k0 + 1];
        }

        v8f Cm = {};   // zero accumulator
        v8f Dm = __builtin_amdgcn_wmma_f32_16x16x4_f32(
            /*neg_a=*/false, A, /*neg_b=*/false, Bm,
            /*c_mod=*/(short)0, Cm, /*reuse_a=*/false, /*reuse_b=*/false);

        // Spill D to LDS. C/D layout: VGPR j -> row M = j (lanes 0-15) or j+8 (16-31),
        // column N = lane&15.
        const int mbase = t * 16 + ((lane < 16) ? 0 : 8);
        #pragma unroll
        for (int j = 0; j < 8; ++j)
            sProj[(mbase + j) * 16 + mlane] = Dm[j];
    }
    __syncthreads();

    // ---------------- Phase 2: gather + bilinear + conf aggregation ----------------
    const int nl  = tid & 63;           // voxel within block
    const int kc  = tid >> 6;           // k-chunk 0..3 (wave-uniform)
    const int n_g = n_base + nl;

    // Reciprocal of per-k confidence sums (normalization over cameras)
    float rsum[8];
    #pragma unroll
    for (int j = 0; j < 8; ++j) {
        const int k = kc * 8 + j;
        float s = 0.0f;
        #pragma unroll
        for (int c = 0; c < CC; ++c) s += conf[(b * CC + c) * KK + k];
        rsum[j] = 1.0f / s;
    }

    float acc[8];
    #pragma unroll
    for (int j = 0; j < 8; ++j) acc[j] = 0.0f;

    #pragma unroll 1   // keep one camera's 32 gathers in flight; bound VGPR pressure
    for (int c = 0; c < CC; ++c) {
        const float px = sProj[nl * 16 + c * 3 + 0];
        const float py = sProj[nl * 16 + c * 3 + 1];
        const float pz = sProj[nl * 16 + c * 3 + 2];

        const bool  valid = (pz > 0.0f);
        const float zs    = (pz == 0.0f) ? 1.0f : pz;
        const float u = px / zs;
        const float v = py / zs;
        // ix = (u/H)*(W-1), iy = (v/W)*(H-1)  (reference's H/W swap preserved)
        const float ix = u * ((float)(WW - 1) / (float)HH);
        const float iy = v * ((float)(HH - 1) / (float)WW);

        const float x0 = floorf(ix), y0 = floorf(iy);
        const float x1 = x0 + 1.0f,  y1 = y0 + 1.0f;
        const float wx1 = ix - x0, wx0 = 1.0f - wx1;
        const float wy1 = iy - y0, wy0 = 1.0f - wy1;

        // fold in-bounds + validity masks into the weights (clamped loads stay legal)
        const float fx0 = (x0 >= 0.0f && x0 <= (float)(WW - 1)) ? wx0 : 0.0f;
        const float fx1 = (x1 >= 0.0f && x1 <= (float)(WW - 1)) ? wx1 : 0.0f;
        const float fy0 = ((y0 >= 0.0f && y0 <= (float)(HH - 1)) && valid) ? wy0 : 0.0f;
        const float fy1 = ((y1 >= 0.0f && y1 <= (float)(HH - 1)) && valid) ? wy1 : 0.0f;

        // clamp to image (NaN-safe: fmaxf/fminf drop NaN operand)
        const int x0i = (int)fminf(fmaxf(x0, 0.0f), (float)(WW - 1));
        const int x1i = (int)fminf(fmaxf(x1, 0.0f), (float)(WW - 1));
        const int y0i = (int)fminf(fmaxf(y0, 0.0f), (float)(HH - 1));
        const int y1i = (int)fminf(fmaxf(y1, 0.0f), (float)(HH - 1));

        // pack the two x-columns into [xb, xb+1]; remap corner weights onto elements
        const int  xb = (x0i < (WW - 2)) ? x0i : (WW - 2);
        const bool a0 = (x0i == xb);
        const bool a1 = (x1i == xb);
        const float w00 = fx0 * fy0, w10 = fx1 * fy0;
        const float w01 = fx0 * fy1, w11 = fx1 * fy1;
        const float e00 = (a0 ? w00 : 0.0f) + (a1 ? w10 : 0.0f);  // (y0, xb)
        const float e01 = (a0 ? 0.0f : w00) + (a1 ? 0.0f : w10);  // (y0, xb+1)
        const float e10 = (a0 ? w01 : 0.0f) + (a1 ? w11 : 0.0f);  // (y1, xb)
        const float e11 = (a0 ? 0.0f : w01) + (a1 ? 0.0f : w11);  // (y1, xb+1)

        const int o0 = y0i * WW + xb;
        const int o1 = y1i * WW + xb;
        const float* hb = heatmaps + (size_t)(b * CC + c) * KK * HW + (size_t)kc * 8 * HW;
        const float* cf = conf + (b * CC + c) * KK + kc * 8;

        float cn[8];
        #pragma unroll
        for (int j = 0; j < 8; ++j) cn[j] = cf[j] * rsum[j];

        #pragma unroll
        for (int j = 0; j < 8; ++j) {
            const float* hk = hb + (size_t)j * HW;
            const float g00 = hk[o0];
            const float g01 = hk[o0 + 1];
            const float g10 = hk[o1];
            const float g11 = hk[o1 + 1];
            const float val = g00 * e00 + g01 * e01 + g10 * e10 + g11 * e11;
            acc[j] = fmaf(val, cn[j], acc[j]);
        }
    }

    // Coalesced stores: lanes are consecutive n for fixed k
    #pragma unroll
    for (int j = 0; j < 8; ++j) {
        const int k = kc * 8 + j;
        out[(size_t)(b * KK + k) * NN + n_g] = acc[j];
    }
}

extern "C" void kernel_launch(void* const* d_in, const int* in_sizes, int n_in,
                              void* d_out, int out_size, void* d_ws, size_t ws_size,
                              hipStream_t stream) {
    (void)in_sizes; (void)n_in; (void)out_size; (void)d_ws; (void)ws_size;
    const float* heatmaps = (const float*)d_in[0];
    const float* Pm       = (const float*)d_in[1];
    const float* coords   = (const float*)d_in[2];
    const float* conf     = (const float*)d_in[3];
    float* out            = (float*)d_out;

    dim3 grid(NN / 64, BB);
    dim3 block(256);
    VolumetricTriangulation_30468497997961_kernel<<<grid, block, 0, stream>>>(
        heatmaps, Pm, coords, conf, out);
}